// DigitCaps_34608846471518
// MI455X (gfx1250) — compile-verified
//
#include <hip/hip_runtime.h>
#include <math.h>

#define B_ 256
#define R_ 192
#define C_ 96
#define O_ 16
#define I_ 20
#define NITER 3

typedef __attribute__((ext_vector_type(2))) float v2f;
typedef __attribute__((ext_vector_type(8))) float v8f;

// ---------------------------------------------------------------------------
// Kernel A: u_hat[r,c,b,o] = sum_i W[r,c,o,i] * x[b,r,i]
// One wave per (r, c, b-tile). GEMM tile M=16 (batch), N=16 (o), K=20 = 5 x 4.
// Uses V_WMMA_F32_16X16X4_F32 (exact fp32).
//
// fp32 A layout (16x4): lane L<16 holds (M=L, K=k0..k0+1) in v0..v1,
// lane L+16 holds (M=L, K=k0+2..k0+3). B (4x16) mirrored. Both map to one
// aligned float2 load per lane per K-chunk.
// ---------------------------------------------------------------------------
__global__ __launch_bounds__(512) void uhat_wmma_kernel(
    const float* __restrict__ x,    // [B, R, I]
    const float* __restrict__ W,    // [R, C, O, I]
    float* __restrict__ uhat)       // [R, C, B, O]
{
    const int rc    = blockIdx.x;          // r*C + c
    const int r     = rc / C_;
    const int wave  = threadIdx.x >> 5;    // 16 waves -> 16 b-tiles
    const int lane  = threadIdx.x & 31;
    const int half  = lane >> 4;           // selects K+0/1 vs K+2/3
    const int lsub  = lane & 15;           // M (A) / N==o (B) index
    const int bbase = wave << 4;

    // B-matrix registers: W[r, c, o=lsub, i = 4j + 2*half .. +1], 5 chunks
    const float* wp = W + ((size_t)rc * O_ + lsub) * I_ + 2 * half;
    v2f bw[5];
#pragma unroll
    for (int j = 0; j < 5; ++j)
        bw[j] = *(const v2f*)(wp + 4 * j);

    // A-matrix: x[b = bbase+lsub, r, i = 4j + 2*half .. +1]
    const float* xp = x + ((size_t)(bbase + lsub) * R_ + r) * I_ + 2 * half;

    v8f acc = {};
#pragma unroll
    for (int j = 0; j < 5; ++j) {
        v2f av = *(const v2f*)(xp + 4 * j);
        acc = __builtin_amdgcn_wmma_f32_16x16x4_f32(
            /*neg_a=*/false, av, /*neg_b=*/false, bw[j],
            /*c_mod=*/(short)0, acc, /*reuse_a=*/false, /*reuse_b=*/false);
    }

    // D layout: VGPR v holds row M = v + 8*half, col N = lsub (= o)
    float* op = uhat + ((size_t)rc * B_ + bbase + half * 8) * O_ + lsub;
#pragma unroll
    for (int v = 0; v < 8; ++v)
        op[(size_t)v * O_] = acc[v];
}

// ---------------------------------------------------------------------------
// Kernel B: c_ij = softmax over r of b_ij (R x C). One block, C threads.
// use_b == 0 -> uniform 1/R (iteration 0; never reads poisoned ws).
// ---------------------------------------------------------------------------
__global__ void softmax_r_kernel(const float* __restrict__ b_ij,
                                 float* __restrict__ c_ij, int use_b)
{
    const int c = threadIdx.x;             // 0..C-1
    if (!use_b) {
        const float u = 1.0f / (float)R_;
        for (int r = 0; r < R_; ++r) c_ij[r * C_ + c] = u;
        return;
    }
    float mx = -INFINITY;
    for (int r = 0; r < R_; ++r) mx = fmaxf(mx, b_ij[r * C_ + c]);
    float sum = 0.0f;
    for (int r = 0; r < R_; ++r) sum += __expf(b_ij[r * C_ + c] - mx);
    const float inv = 1.0f / sum;
    for (int r = 0; r < R_; ++r)
        c_ij[r * C_ + c] = __expf(b_ij[r * C_ + c] - mx) * inv;
}

// ---------------------------------------------------------------------------
// Kernel C: s_j[b,c,o] = sum_r c_ij[r,c] * u_hat[r,c,b,o]; then squash.
// Block = one (c, 16-batch chunk); tid = (b_local<<4) | o. Fully coalesced
// 1KB reads per r. Squash via 16-lane shfl_xor reduction over o.
// ---------------------------------------------------------------------------
__global__ __launch_bounds__(256) void route_squash_kernel(
    const float* __restrict__ c_ij,   // [R, C]
    const float* __restrict__ uhat,   // [R, C, B, O]
    float* __restrict__ vout)         // [B, C, O]
{
    __shared__ float cs[R_];
    const int c      = blockIdx.x >> 4;
    const int bchunk = blockIdx.x & 15;
    const int tid    = threadIdx.x;

    if (tid < R_) cs[tid] = c_ij[tid * C_ + c];
    __syncthreads();

    const int o = tid & 15;
    const int b = (bchunk << 4) + (tid >> 4);

    const float* up = uhat + ((size_t)c * B_ + b) * O_ + o;
    float s = 0.0f;
#pragma unroll 4
    for (int r = 0; r < R_; ++r)
        s += cs[r] * up[(size_t)r * ((size_t)C_ * B_ * O_)];

    float n2 = s * s;
    n2 += __shfl_xor(n2, 1, 32);
    n2 += __shfl_xor(n2, 2, 32);
    n2 += __shfl_xor(n2, 4, 32);
    n2 += __shfl_xor(n2, 8, 32);

    const float scale = sqrtf(n2) / (1.0f + n2);   // (n2/(1+n2)) * 1/||s||
    vout[((size_t)b * C_ + c) * O_ + o] = s * scale;
}

// ---------------------------------------------------------------------------
// Kernel D: a_ij[r,c] = (1/B) sum_{b,o} u_hat[r,c,b,o] * v[b,c,o];
//           b_ij = a (first) or b_ij += a. Block = one (r,c); 4096-elem
// contiguous reduction of u_hat.
// ---------------------------------------------------------------------------
__global__ __launch_bounds__(256) void agreement_kernel(
    const float* __restrict__ uhat,   // [R, C, B, O]
    const float* __restrict__ v,      // [B, C, O]
    float* __restrict__ b_ij,         // [R, C]
    int first)
{
    const int rc  = blockIdx.x;
    const int c   = rc % C_;
    const int tid = threadIdx.x;

    const float* up = uhat + (size_t)rc * B_ * O_;
    float p = 0.0f;
#pragma unroll
    for (int k = 0; k < (B_ * O_) / 256; ++k) {     // 16 steps
        const int idx = k * 256 + tid;
        const int b = idx >> 4, o = idx & 15;
        p += up[idx] * v[((size_t)b * C_ + c) * O_ + o];
    }
    p += __shfl_xor(p, 1, 32);
    p += __shfl_xor(p, 2, 32);
    p += __shfl_xor(p, 4, 32);
    p += __shfl_xor(p, 8, 32);
    p += __shfl_xor(p, 16, 32);

    __shared__ float red[8];
    const int lane = tid & 31, wv = tid >> 5;
    if (lane == 0) red[wv] = p;
    __syncthreads();
    if (tid == 0) {
        float t = 0.0f;
#pragma unroll
        for (int i = 0; i < 8; ++i) t += red[i];
        t *= (1.0f / (float)B_);
        b_ij[rc] = first ? t : (b_ij[rc] + t);
    }
}

// ---------------------------------------------------------------------------
extern "C" void kernel_launch(void* const* d_in, const int* in_sizes, int n_in,
                              void* d_out, int out_size, void* d_ws, size_t ws_size,
                              hipStream_t stream)
{
    const float* x = (const float*)d_in[0];   // [B, R, I]
    const float* W = (const float*)d_in[1];   // [R, C, O, I]
    float* out = (float*)d_out;               // [B, C, O]

    float* ws   = (float*)d_ws;
    float* uhat = ws;                                       // R*C*B*O floats
    float* b_ij = uhat + (size_t)R_ * C_ * B_ * O_;         // R*C
    float* c_ij = b_ij + (size_t)R_ * C_;                   // R*C
    float* v_j  = c_ij + (size_t)R_ * C_;                   // B*C*O

    // Stage 1: u_hat via fp32 WMMA. One block per (r,c), 16 waves = 16 b-tiles.
    uhat_wmma_kernel<<<R_ * C_, 512, 0, stream>>>(x, W, uhat);

    // Stage 2: dynamic routing, 3 iterations.
    for (int it = 0; it < NITER; ++it) {
        softmax_r_kernel<<<1, C_, 0, stream>>>(b_ij, c_ij, it > 0 ? 1 : 0);
        float* vo = (it == NITER - 1) ? out : v_j;
        route_squash_kernel<<<C_ * (B_ / 16), 256, 0, stream>>>(c_ij, uhat, vo);
        if (it < NITER - 1)
            agreement_kernel<<<R_ * C_, 256, 0, stream>>>(uhat, v_j, b_ij,
                                                          it == 0 ? 1 : 0);
    }
}